// Postprocess_18339510354491
// MI455X (gfx1250) — compile-verified
//
#include <hip/hip_runtime.h>
#include <stdint.h>

// ---------------------------------------------------------------------------
// Human3.6M pose post-process, MI455X (gfx1250, wave32).
//
// Roofline: ~172 MB traffic -> ~7.4us floor at 23.3 TB/s. Round-1 kernel was
// issue-bound (~7K static instrs, mostly precise OCML sincos). This round
// switches to native v_sin_f32/v_cos_f32 (__sincosf) to get the instruction
// stream well under the memory floor.
//
// Data path:
//   * per wave: async-copy 32 pred rows (32*264B = 8448B = 528 x B128) from
//     global into LDS via global_load_async_to_lds_b128 (ASYNCcnt), then
//     s_wait_asynccnt 0 (per-wave LDS partition -> no workgroup barrier).
//   * each lane consumes its own 66-float row from LDS (stride 66 floats ->
//     bank = 2*lane + k, conflict-free across wave32), computes sincos +
//     kinematic prefix-sum fully in registers, stores 96 floats as 24 x b128.
// ---------------------------------------------------------------------------

#define BLOCK_THREADS   128          // 4 waves of 32
#define ROWS_PER_BLOCK  128          // 1 item per thread
#define ROW_FLOATS      66
#define ROW_BYTES       (ROW_FLOATS * 4)            // 264
#define WAVE_ROWS       32
#define WAVE_BYTES      (WAVE_ROWS * ROW_BYTES)     // 8448
#define WAVE_CHUNKS     (WAVE_BYTES / 16)           // 528 x b128
#define T_PRED          64
#define T_OBS           16
#define OBS_ROW         96
#define OUT_ROW         96

__global__ __launch_bounds__(BLOCK_THREADS)
void pose_post_kernel(const float* __restrict__ obs,    // (B, 16, 96)
                      const float* __restrict__ pred,   // (B, 64, 66)
                      float* __restrict__ out,          // (B, 64, 96)
                      int n_items)                      // B * 64
{
    __shared__ __align__(16) float smem[ROWS_PER_BLOCK * ROW_FLOATS];

    const int tid   = threadIdx.x;
    const int lane  = tid & 31;
    const int wave  = tid >> 5;
    const int block_row0 = blockIdx.x * ROWS_PER_BLOCK;
    const int wave_row0  = block_row0 + wave * WAVE_ROWS;
    const int item       = block_row0 + tid;

    // ---- stage this wave's 32 pred rows into LDS -------------------------
    if (wave_row0 + WAVE_ROWS <= n_items) {
        // full wave: 528 contiguous 16B chunks, async global -> LDS
        const uint64_t gbase = (uint64_t)(uintptr_t)pred
                             + (uint64_t)wave_row0 * ROW_BYTES;
        const uint32_t lbase = (uint32_t)(uintptr_t)
                               (&smem[wave * WAVE_ROWS * ROW_FLOATS]);
        #pragma unroll
        for (int j = 0; j < 17; ++j) {
            int c = j * 32 + lane;                 // chunk index
            if (c < WAVE_CHUNKS) {
                uint64_t ga = gbase + (uint32_t)c * 16u;
                uint32_t la = lbase + (uint32_t)c * 16u;
                asm volatile("global_load_async_to_lds_b128 %0, %1, off"
                             :: "v"(la), "v"(ga) : "memory");
            }
        }
    } else {
        // tail (doesn't occur for the fixed 4096x64 shape): plain copy
        if (item < n_items) {
            const float* g = pred + (size_t)item * ROW_FLOATS;
            for (int k = 0; k < ROW_FLOATS; ++k)
                smem[tid * ROW_FLOATS + k] = g[k];
        }
    }
    // wait for this wave's async transfers
    asm volatile("s_wait_asynccnt 0x0" ::: "memory");

    if (item >= n_items) return;

    const float* __restrict__ row = &smem[tid * ROW_FLOATS];

    // kinematic tree (topologically ordered), compile-time constants
    const int CHD[22] = {12,13,14,15,25,26,27,29,30,17,18,19,21,22,
                          2, 3, 4, 5, 7, 8, 9,10};
    const int PAR[22] = {11,12,13,14,13,25,26,27,29,13,17,18,19,21,
                          1, 2, 3, 4, 6, 7, 8, 9};

    float X[96];

    // spherical -> xyz into child joints: (r*sin(phi)*cos(th), r*cos(phi),
    //                                      r*sin(phi)*sin(th))
    // __sincosf -> native v_sin_f32/v_cos_f32 (inputs are ~N(0,1), well
    // inside the hw units' accurate range; avoids libm's unrolled
    // Payne-Hanek bloat that made round 1 issue-bound).
    #pragma unroll
    for (int j = 0; j < 22; ++j) {
        float r  = row[3 * j + 0];
        float th = row[3 * j + 1];
        float ph = row[3 * j + 2];
        float sth, cth, sph, cph;
        __sincosf(th, &sth, &cth);
        __sincosf(ph, &sph, &cph);
        float rs = r * sph;
        X[CHD[j] * 3 + 0] = rs * cth;
        X[CHD[j] * 3 + 1] = r * cph;
        X[CHD[j] * 3 + 2] = rs * sth;
    }

    // root joints copied from LAST observed frame (uniform per 64 items;
    // tiny working set, lives in L2/WGP$)
    {
        const int RC[4] = {0, 1, 6, 11};
        const int b = item / T_PRED;
        const float* ob = obs + ((size_t)b * T_OBS + (T_OBS - 1)) * OBS_ROW;
        #pragma unroll
        for (int k = 0; k < 4; ++k) {
            X[RC[k] * 3 + 0] = ob[RC[k] * 3 + 0];
            X[RC[k] * 3 + 1] = ob[RC[k] * 3 + 1];
            X[RC[k] * 3 + 2] = ob[RC[k] * 3 + 2];
        }
    }

    // sequential prefix-sum along the tree (parents final before children)
    #pragma unroll
    for (int j = 0; j < 22; ++j) {
        X[CHD[j] * 3 + 0] += X[PAR[j] * 3 + 0];
        X[CHD[j] * 3 + 1] += X[PAR[j] * 3 + 1];
        X[CHD[j] * 3 + 2] += X[PAR[j] * 3 + 2];
    }

    // x[IDX_IGNORE] = x[IDX_EQUAL]
    {
        const int IG[6] = {16, 20, 23, 24, 28, 31};
        const int EQ[6] = {13, 19, 22, 13, 27, 30};
        #pragma unroll
        for (int k = 0; k < 6; ++k) {
            X[IG[k] * 3 + 0] = X[EQ[k] * 3 + 0];
            X[IG[k] * 3 + 1] = X[EQ[k] * 3 + 1];
            X[IG[k] * 3 + 2] = X[EQ[k] * 3 + 2];
        }
    }

    // store 96 floats as 24 x b128 (row base = item*384B, 16B aligned)
    float4* o4 = (float4*)(out + (size_t)item * OUT_ROW);
    #pragma unroll
    for (int q = 0; q < 24; ++q) {
        o4[q] = make_float4(X[4 * q + 0], X[4 * q + 1],
                            X[4 * q + 2], X[4 * q + 3]);
    }
}

extern "C" void kernel_launch(void* const* d_in, const int* in_sizes, int n_in,
                              void* d_out, int out_size, void* d_ws, size_t ws_size,
                              hipStream_t stream) {
    const float* obs  = (const float*)d_in[0];   // (4096, 16, 96) f32
    const float* pred = (const float*)d_in[1];   // (4096, 64, 66) f32
    float* out        = (float*)d_out;           // (4096, 64, 96) f32

    const int n_items = in_sizes[1] / ROW_FLOATS;         // B*T = 262144
    const int blocks  = (n_items + ROWS_PER_BLOCK - 1) / ROWS_PER_BLOCK;

    pose_post_kernel<<<blocks, BLOCK_THREADS, 0, stream>>>(obs, pred, out,
                                                           n_items);
}